// OverlapSamplingModule_61546881352069
// MI455X (gfx1250) — compile-verified
//
#include <hip/hip_runtime.h>
#include <stdint.h>

typedef __attribute__((ext_vector_type(2))) float v2f;
typedef __attribute__((ext_vector_type(8))) float v8f;

#define BC      8
#define NP      4096
#define M1      1024
#define MBASE   (BC * M1)          // 8192
#define MEXTRA  1638
#define MQ      (MBASE + MEXTRA)   // 9830
#define KNN     16
#define TILES_PER_CLOUD 116        // ceil((1024+820)/16)
#define CHUNK   512
#define NCHUNK  (NP / CHUNK)

// ---------------------------------------------------------------------------
// Stage 1: farthest point sampling, one workgroup per cloud.
// 1024 threads, 4 points per thread held in registers. Greedy FPS with
// deterministic start at 0; argmax with lowest-index tie-break via packed
// u64 keys (non-negative f32 bit pattern orders like the float).
// ---------------------------------------------------------------------------
__global__ __launch_bounds__(1024)
void fps_kernel(const float* __restrict__ pos, int* __restrict__ picks)
{
    const int b   = blockIdx.x;
    const int tid = threadIdx.x;
    const float* p = pos + (size_t)b * NP * 3;

    __shared__ unsigned long long s_part[32];
    __shared__ int   s_last;
    __shared__ float s_cx, s_cy, s_cz;

    float px[4], py[4], pz[4], mind[4];
#pragma unroll
    for (int k = 0; k < 4; ++k) {
        int i = tid + k * 1024;
        px[k] = p[i * 3 + 0];
        py[k] = p[i * 3 + 1];
        pz[k] = p[i * 3 + 2];
        mind[k] = __builtin_inff();
    }

    if (tid == 0) { picks[b * M1] = 0; s_last = 0; }
    __syncthreads();

    for (int t = 1; t < M1; ++t) {
        const int last = s_last;
        if (tid == (last & 1023)) {
            const int kk = last >> 10;
#pragma unroll
            for (int k = 0; k < 4; ++k)
                if (k == kk) { s_cx = px[k]; s_cy = py[k]; s_cz = pz[k]; }
        }
        __syncthreads();

        const float cx = s_cx, cy = s_cy, cz = s_cz;
        unsigned long long best = 0ull;
#pragma unroll
        for (int k = 0; k < 4; ++k) {
            const float dx = px[k] - cx, dy = py[k] - cy, dz = pz[k] - cz;
            const float d  = dx * dx + dy * dy + dz * dz;
            mind[k] = fminf(mind[k], d);
            const unsigned int vb  = __float_as_uint(mind[k]);
            const unsigned int idx = (unsigned int)(tid + k * 1024);
            const unsigned long long key =
                ((unsigned long long)vb << 32) |
                (unsigned long long)(0xFFFFFFFFu - idx);
            best = key > best ? key : best;
        }
        // wave32 reduction
#pragma unroll
        for (int off = 16; off > 0; off >>= 1) {
            unsigned long long o = __shfl_xor(best, off, 32);
            best = o > best ? o : best;
        }
        if ((tid & 31) == 0) s_part[tid >> 5] = best;
        __syncthreads();
        if (tid < 32) {
            unsigned long long v = s_part[tid];
#pragma unroll
            for (int off = 16; off > 0; off >>= 1) {
                unsigned long long o = __shfl_xor(v, off, 32);
                v = o > v ? o : v;
            }
            if (tid == 0) {
                const int idx = (int)(0xFFFFFFFFu - (unsigned int)(v & 0xFFFFFFFFull));
                s_last = idx;
                picks[b * M1 + t] = idx;
            }
        }
        __syncthreads();
    }
}

// ---------------------------------------------------------------------------
// Stage 2: batched kNN via V_WMMA_F32_16X16X4_F32.
//   A row m  = (-2qx, -2qy, -2qz, 1)
//   B col n  = ( px,   py,   pz,  |p|^2)   <- column norm folded into B (K=3)
//   C[m][n]  = |q|^2[m]                    <- row-only => loop-invariant v8f
//   D[m][n]  = full squared distance.
// C is built once (8 LDS reads), so the inner loop carries no accumulator
// splat; each 16x16 distance tile = 3 loads + few VALU + 1 WMMA + 8 DS stores.
// Next point-chunk is prefetched (global_prefetch_b8) while this chunk's
// WMMAs and the LDS top-K scan run.
// ---------------------------------------------------------------------------
__global__ __launch_bounds__(256)
void knn_kernel(const float* __restrict__ pos, const int* __restrict__ picks,
                int* __restrict__ out)
{
    const int blk   = blockIdx.x;
    const int b     = blk / TILES_PER_CLOUD;
    const int tile  = blk % TILES_PER_CLOUD;
    const int extra = (b == 0) ? 820 : (b == 1) ? 818 : 0;
    const int qcnt  = M1 + extra;
    const int lq0   = tile * 16;
    if (lq0 >= qcnt) return;               // uniform per block

    const int  tid  = threadIdx.x;
    const int  lane = tid & 31;
    const int  wave = tid >> 5;
    const int  m    = lane & 15;
    const bool hi   = lane >= 16;

    __shared__ float sA0[16], sA1[16], sA2[16], sA3[16];
    __shared__ int   s_row[16];
    __shared__ float s_dist[16][CHUNK];
    __shared__ float s_topd[16][KNN];
    __shared__ int   s_topi[16][KNN];

    if (tid < 16) {
        const int  lq    = lq0 + tid;
        const bool valid = lq < qcnt;
        const int  lql   = valid ? lq : 0;
        const int  pk    = picks[b * M1 + (lql < M1 ? lql : lql - M1)];
        int row;
        if (!valid)          row = -1;
        else if (lq < M1)    row = b * M1 + lq;
        else                 row = MBASE + ((b == 0) ? 0 : 820) + (lq - M1);

        const float qx = pos[((size_t)(b * NP + pk)) * 3 + 0];
        const float qy = pos[((size_t)(b * NP + pk)) * 3 + 1];
        const float qz = pos[((size_t)(b * NP + pk)) * 3 + 2];
        sA0[tid] = -2.0f * qx;
        sA1[tid] = -2.0f * qy;
        sA2[tid] = -2.0f * qz;
        sA3[tid] = qx * qx + qy * qy + qz * qz;   // |q|^2
        s_row[tid] = row;
        if (valid) out[row] = b * NP + pk;        // combined_idx
#pragma unroll
        for (int k2 = 0; k2 < KNN; ++k2) {
            s_topd[tid][k2] = __builtin_inff();
            s_topi[tid][k2] = 0;
        }
    }
    __syncthreads();

    // loop-invariant A operand for this lane: (K0,K1)=( -2qx,-2qy ), (K2,K3)=( -2qz, 1 )
    v2f a;
    a.x = hi ? sA2[m] : sA0[m];
    a.y = hi ? 1.0f   : sA1[m];

    // loop-invariant C accumulator: C[m][n] = |q_m|^2 (row-only)
    v8f cacc;
#pragma unroll
    for (int j = 0; j < 8; ++j) cacc[j] = sA3[j + (hi ? 8 : 0)];

    const float* pcl = pos + (size_t)(b * NP) * 3;

    for (int ch = 0; ch < NCHUNK; ++ch) {
        const int base = ch * CHUNK + wave * 64;

        // prefetch next chunk's slice for this wave (cacheline granular,
        // per-lane addresses stride 2 points = 24B to cover the 768B slice)
        if (ch + 1 < NCHUNK) {
            const int pn = base + CHUNK + lane * 2;
            __builtin_prefetch(&pcl[(size_t)pn * 3], 0, 3);
        }

#pragma unroll
        for (int t = 0; t < 4; ++t) {
            const int n = base + t * 16 + m;        // point index in cloud
            const float px_ = pcl[n * 3 + 0];
            const float py_ = pcl[n * 3 + 1];
            const float pz_ = pcl[n * 3 + 2];
            const float pn2 = px_ * px_ + py_ * py_ + pz_ * pz_;
            v2f bm;
            bm.x = hi ? pz_ : px_;
            bm.y = hi ? pn2 : py_;

            // D = A x B + C : 16x16 tile of squared distances
            v8f dmat = __builtin_amdgcn_wmma_f32_16x16x4_f32(
                false, a, false, bm, (short)0, cacc, false, false);

            const int lcol = wave * 64 + t * 16 + m;    // 0..511
#pragma unroll
            for (int j = 0; j < 8; ++j)
                s_dist[j + (hi ? 8 : 0)][lcol] = dmat[j];
        }
        __syncthreads();

        if (tid < 16) {
            const int r     = tid;
            const int cbase = ch * CHUNK;
            for (int cidx = 0; cidx < CHUNK; ++cidx) {
                const float d = s_dist[r][cidx];
                if (d < s_topd[r][KNN - 1]) {
                    int j = KNN - 1;
                    while (j > 0 && s_topd[r][j - 1] > d) {
                        s_topd[r][j] = s_topd[r][j - 1];
                        s_topi[r][j] = s_topi[r][j - 1];
                        --j;
                    }
                    s_topd[r][j] = d;
                    s_topi[r][j] = cbase + cidx;
                }
            }
        }
        __syncthreads();
    }

    if (tid < 16) {
        const int row = s_row[tid];
        if (row >= 0) {
            int* rowArr = out + MQ;
            int* colArr = out + MQ + MQ * KNN;
#pragma unroll
            for (int k2 = 0; k2 < KNN; ++k2) {
                rowArr[row * KNN + k2] = row;
                colArr[row * KNN + k2] = b * NP + s_topi[tid][k2];
            }
        }
    }
}

extern "C" void kernel_launch(void* const* d_in, const int* in_sizes, int n_in,
                              void* d_out, int out_size, void* d_ws, size_t ws_size,
                              hipStream_t stream) {
    (void)in_sizes; (void)n_in; (void)out_size; (void)ws_size;
    const float* pos  = (const float*)d_in[0];
    // d_in[1] (batch) is implied by the fixed equal-sized-cloud layout.
    int* picks = (int*)d_ws;                 // B * M1 int32 scratch
    int* out   = (int*)d_out;                // [combined_idx | row | col] int32

    fps_kernel<<<BC, 1024, 0, stream>>>(pos, picks);
    knn_kernel<<<BC * TILES_PER_CLOUD, 256, 0, stream>>>(pos, picks, out);
}